// MyGNN_13460427505824
// MI455X (gfx1250) — compile-verified
//
#include <hip/hip_runtime.h>

// ---------------------------------------------------------------------------
// GCN forward on gfx1250 (MI455X).
//  - Dense GEMMs via V_WMMA_F32_16X16X4_F32 (fp32, matches reference math).
//  - Edge aggregation: (M/4)-lane-per-edge float4 coalesced gather + f32 global atomics
//    (accumulator is L2-resident: <=25.6MB << 192MB L2).
//  - BN stats via LDS ds_add_f32 block reduction -> global atomics.
//  - Max pool via bit-monotonic atomicMax (values >= 0 after ReLU).
// ---------------------------------------------------------------------------

typedef __attribute__((ext_vector_type(2))) float v2f;
typedef __attribute__((ext_vector_type(8))) float v8f;

#define BN_EPS 1e-5f

__global__ void zero_f32(float* __restrict__ p, long long n) {
  long long i = (long long)blockIdx.x * blockDim.x + threadIdx.x;
  if (i < n) p[i] = 0.0f;
}

__global__ void deg_kernel(const int* __restrict__ ei, float* __restrict__ deg, int E) {
  int e = blockIdx.x * blockDim.x + threadIdx.x;
  if (e < E) atomicAdd(&deg[ei[E + e]], 1.0f);
}

__global__ void dinv_kernel(float* __restrict__ d, int N) {
  int i = blockIdx.x * blockDim.x + threadIdx.x;
  if (i < N) d[i] = rsqrtf(d[i] + 1.0f);  // +1 self loop
}

// H[N x M] = A[N x K] @ W[K x M], via 16x16 output tiles per wave.
// ISA 7.12.2 layouts:
//   A (16x4 f32):  lanes 0-15 hold K=k0+{0,1}, lanes 16-31 hold K=k0+{2,3}, row = lane%16
//   B (4x16 f32):  vgpr j: lanes 0-15 = row k0+j, lanes 16-31 = row k0+2+j, col = lane%16
//   C/D (16x16):   vgpr v: lanes 0-15 -> row v, lanes 16-31 -> row v+8, col = lane%16
__global__ void gemm_wmma_f32(const float* __restrict__ A, const float* __restrict__ W,
                              float* __restrict__ H, int N, int K, int M) {
  int wave = (int)(((long long)blockIdx.x * blockDim.x + threadIdx.x) >> 5);
  int lane = threadIdx.x & 31;
  int colTiles = M >> 4;
  int rowTile = wave / colTiles;
  int colTile = wave - rowTile * colTiles;
  if (rowTile * 16 >= N) return;            // wave-uniform: EXEC stays all-ones
  int r0 = rowTile << 4, c0 = colTile << 4;
  int lrow = lane & 15;
  int hi = lane >> 4;                       // 0: lanes 0-15, 1: lanes 16-31

  v8f acc = {};
  const float* arow = A + (size_t)(r0 + lrow) * K + (hi << 1);
  const float* bcol = W + (size_t)(hi << 1) * M + c0 + lrow;
#pragma unroll 4
  for (int k0 = 0; k0 < K; k0 += 4) {
    v2f a; a.x = arow[k0];            a.y = arow[k0 + 1];
    v2f b; b.x = bcol[(size_t)k0 * M]; b.y = bcol[(size_t)(k0 + 1) * M];
    acc = __builtin_amdgcn_wmma_f32_16x16x4_f32(
        /*neg_a=*/false, a, /*neg_b=*/false, b,
        /*c_mod=*/(short)0, acc, /*reuse_a=*/false, /*reuse_b=*/false);
  }
  float* out = H + (size_t)(r0 + (hi << 3)) * M + c0 + lrow;
#pragma unroll
  for (int v = 0; v < 8; ++v) out[(size_t)v * M] = acc[v];
}

// agg[dst] += h[src] * dinv[src]*dinv[dst].  (M/4) lanes per edge, float4 gather.
__global__ void edge_agg(const int* __restrict__ ei, const float* __restrict__ h,
                         const float* __restrict__ dinv, float* __restrict__ agg,
                         int E, int M, int chunksLog2) {
  long long tid = (long long)blockIdx.x * blockDim.x + threadIdx.x;
  int chunks = M >> 2;
  if (tid >= (long long)E * chunks) return;
  int e = (int)(tid >> chunksLog2);
  int c = (int)(tid & (chunks - 1));
  int s = ei[e];
  int d = ei[E + e];
  float coef = dinv[s] * dinv[d];
  const float4 v = ((const float4*)(h + (size_t)s * M))[c];
  float* ad = agg + (size_t)d * M + (c << 2);
  atomicAdd(ad + 0, v.x * coef);
  atomicAdd(ad + 1, v.y * coef);
  atomicAdd(ad + 2, v.z * coef);
  atomicAdd(ad + 3, v.w * coef);
}

// z = agg + h*dinv^2 + bias (in place into agg); accumulate per-feature sum/sumsq.
__global__ void node_finalize(float* __restrict__ agg, const float* __restrict__ h,
                              const float* __restrict__ dinv, const float* __restrict__ bias,
                              float* __restrict__ sum, float* __restrict__ sumsq,
                              int N, int M, int mLog2) {
  __shared__ float s_sum[64];
  __shared__ float s_sq[64];
  int t = threadIdx.x;
  if (t < M) { s_sum[t] = 0.0f; s_sq[t] = 0.0f; }
  __syncthreads();
  long long tid = (long long)blockIdx.x * blockDim.x + t;
  if (tid < (long long)N * M) {
    int i = (int)(tid >> mLog2);
    int f = (int)(tid & (M - 1));
    float di = dinv[i];
    float z = agg[tid] + h[tid] * di * di + bias[f];
    agg[tid] = z;
    atomicAdd(&s_sum[f], z);
    atomicAdd(&s_sq[f], z * z);
  }
  __syncthreads();
  if (t < M) {
    atomicAdd(&sum[t], s_sum[t]);
    atomicAdd(&sumsq[t], s_sq[t]);
  }
}

__global__ void bn_finalize(const float* __restrict__ sum, const float* __restrict__ sumsq,
                            const float* __restrict__ g, const float* __restrict__ beta,
                            float* __restrict__ scale, float* __restrict__ shift,
                            int N, int M) {
  int f = threadIdx.x;
  if (f < M) {
    float invN = 1.0f / (float)N;
    float mean = sum[f] * invN;
    float var = sumsq[f] * invN - mean * mean;
    float s = g[f] * rsqrtf(var + BN_EPS);
    scale[f] = s;
    shift[f] = beta[f] - mean * s;
  }
}

__global__ void bn_relu(const float* __restrict__ z, const float* __restrict__ scale,
                        const float* __restrict__ shift, float* __restrict__ out,
                        long long total, int M) {
  long long tid = (long long)blockIdx.x * blockDim.x + threadIdx.x;
  if (tid < total) {
    int f = (int)(tid & (M - 1));
    out[tid] = fmaxf(z[tid] * scale[f] + shift[f], 0.0f);
  }
}

// atomicMax on int bits is order-preserving for non-negative floats (post-ReLU).
__global__ void max_pool(const float* __restrict__ act, float* __restrict__ pooled,
                         long long total, int M) {
  long long tid = (long long)blockIdx.x * blockDim.x + threadIdx.x;
  if (tid < total) {
    int f = (int)(tid & (M - 1));
    atomicMax((int*)&pooled[f], __float_as_int(act[tid]));
  }
}

__global__ void out_gemm(const float* __restrict__ pooled, const float* __restrict__ Wout,
                         const float* __restrict__ bout, float* __restrict__ out) {
  int j = threadIdx.x;  // 32 threads
  float acc = bout[j];
#pragma unroll
  for (int k = 0; k < 64; ++k) acc += pooled[k] * Wout[k * 32 + j];
  out[j] = acc;
}

static inline int cdiv_ll(long long a, int b) { return (int)((a + b - 1) / b); }

static void run_layer(const float* act, int K, const float* W, const float* b,
                      const float* g, const float* beta, int M,
                      float* h, float* agg, float* outAct,
                      const float* dinv, float* sum, float* sumsq,
                      float* scale, float* shift,
                      int N, int E, const int* ei, hipStream_t stream) {
  // GEMM h = act @ W
  {
    long long waves = (long long)(N / 16) * (M / 16);
    gemm_wmma_f32<<<cdiv_ll(waves * 32, 256), 256, 0, stream>>>(act, W, h, N, K, M);
  }
  long long nm = (long long)N * M;
  zero_f32<<<cdiv_ll(nm, 256), 256, 0, stream>>>(agg, nm);
  zero_f32<<<1, 256, 0, stream>>>(sum, 128);  // sum + sumsq contiguous
  {
    int chunksLog2 = (M == 16) ? 2 : 4;
    long long tot = (long long)E * (M >> 2);
    edge_agg<<<cdiv_ll(tot, 256), 256, 0, stream>>>(ei, h, dinv, agg, E, M, chunksLog2);
  }
  int mLog2 = (M == 16) ? 4 : 6;
  node_finalize<<<cdiv_ll(nm, 256), 256, 0, stream>>>(agg, h, dinv, b, sum, sumsq, N, M, mLog2);
  bn_finalize<<<1, 64, 0, stream>>>(sum, sumsq, g, beta, scale, shift, N, M);
  bn_relu<<<cdiv_ll(nm, 256), 256, 0, stream>>>(agg, scale, shift, outAct, nm, M);
}

extern "C" void kernel_launch(void* const* d_in, const int* in_sizes, int n_in,
                              void* d_out, int out_size, void* d_ws, size_t ws_size,
                              hipStream_t stream) {
  const float* x    = (const float*)d_in[0];
  const int*   ei   = (const int*)d_in[1];   // int32 per harness convention
  const float* W1   = (const float*)d_in[2];
  const float* b1   = (const float*)d_in[3];
  const float* g1   = (const float*)d_in[4];
  const float* be1  = (const float*)d_in[5];
  const float* W2   = (const float*)d_in[6];
  const float* b2   = (const float*)d_in[7];
  const float* g2   = (const float*)d_in[8];
  const float* be2  = (const float*)d_in[9];
  const float* W3   = (const float*)d_in[10];
  const float* b3   = (const float*)d_in[11];
  const float* g3   = (const float*)d_in[12];
  const float* be3  = (const float*)d_in[13];
  const float* Wout = (const float*)d_in[14];
  const float* bout = (const float*)d_in[15];
  float* out = (float*)d_out;

  const int N = in_sizes[0] / 128;   // 100000
  const int E = in_sizes[1] / 2;     // 3200000

  // Workspace layout (floats):
  float* ws     = (float*)d_ws;
  float* dinv   = ws;                             // N
  float* bufA   = dinv + N;                       // N*64
  float* bufB   = bufA + (size_t)N * 64;          // N*64
  float* stats  = bufB + (size_t)N * 64;          // 256
  float* pooled = stats + 256;                    // 64
  float* sum = stats, * sumsq = stats + 64, * scale = stats + 128, * shift = stats + 192;

  // Degrees -> dinv = rsqrt(deg + 1)
  zero_f32<<<cdiv_ll(N, 256), 256, 0, stream>>>(dinv, N);
  deg_kernel<<<cdiv_ll(E, 256), 256, 0, stream>>>(ei, dinv, E);
  dinv_kernel<<<cdiv_ll(N, 256), 256, 0, stream>>>(dinv, N);

  // Layer 1: x[N,128] -> act1[N,16]   (h=bufA, agg/z=bufB, act->bufA)
  run_layer(x, 128, W1, b1, g1, be1, 16, bufA, bufB, bufA,
            dinv, sum, sumsq, scale, shift, N, E, ei, stream);
  // Layer 2: act1[N,16] -> act2[N,64] (h=bufB, agg/z=bufA, act->bufB)
  run_layer(bufA, 16, W2, b2, g2, be2, 64, bufB, bufA, bufB,
            dinv, sum, sumsq, scale, shift, N, E, ei, stream);
  // Layer 3: act2[N,64] -> act3[N,64] (h=bufA, agg/z=bufB, act->bufA)
  run_layer(bufB, 64, W3, b3, g3, be3, 64, bufA, bufB, bufA,
            dinv, sum, sumsq, scale, shift, N, E, ei, stream);

  // Global max pool (act3 >= 0) then [1,64] @ [64,32] + bout
  zero_f32<<<1, 64, 0, stream>>>(pooled, 64);
  long long nm = (long long)N * 64;
  max_pool<<<cdiv_ll(nm, 256), 256, 0, stream>>>(bufA, pooled, nm, 64);
  out_gemm<<<1, 32, 0, stream>>>(pooled, Wout, bout, out);
}